// MultiHeadGraphAttention_63823214019204
// MI455X (gfx1250) — compile-verified
//
#include <hip/hip_runtime.h>
#include <hip/hip_bf16.h>

// MI455X / gfx1250, wave32. Multi-head graph attention (GAT), eval mode.
// N=4096, IN_DIM=256, HEADS=4, OUT_DIM=64.

#define N_NODES 4096
#define IN_DIM  256
#define HEADS   4
#define OUT_DIM 64
#define WH_COLS (HEADS * OUT_DIM)   // 256

typedef __attribute__((ext_vector_type(2))) float v2f;
typedef __attribute__((ext_vector_type(4))) float v4f;
typedef __attribute__((ext_vector_type(8))) float v8f;

__device__ __forceinline__ float waveReduceSum(float v) {
    #pragma unroll
    for (int off = 16; off > 0; off >>= 1)
        v += __shfl_xor(v, off, 32);
    return v;
}

// LDS byte offset of a __shared__ object: flat->LDS uses addr[31:0] (ISA 10.2).
__device__ __forceinline__ unsigned ldsOff(const void* p) {
    return (unsigned)(uintptr_t)p;
}

// ---------------------------------------------------------------------------
// Kernel 1: Wh = h @ W   (4096x256) = (4096x256)(256x256), fp32 WMMA 16x16x4.
// One wave per 16(M) x 64(N) tile: 4 accumulators, K-loop in steps of 4.
// ---------------------------------------------------------------------------
__global__ __launch_bounds__(32) void gemm_wh(const float* __restrict__ hmat,
                                              const float* __restrict__ W,
                                              float* __restrict__ Wh) {
    const int lane = threadIdx.x;
    const int hlf  = lane >> 4;     // 0/1: which K-pair this lane holds
    const int l16  = lane & 15;
    const int m0 = (blockIdx.x >> 2) * 16;  // 256 row tiles
    const int n0 = (blockIdx.x & 3) * 64;   // 4 column groups of 64

    v8f acc[4] = {};
    const int row = m0 + l16;               // A: lanes 0-15 and 16-31 both map M=0..15

    for (int k = 0; k < IN_DIM; k += 4) {
        const int kk = k + 2 * hlf;         // lane half selects K={0,1} or K={2,3}
        v2f afrag;
        afrag.x = hmat[row * IN_DIM + kk];
        afrag.y = hmat[row * IN_DIM + kk + 1];
        #pragma unroll
        for (int c = 0; c < 4; ++c) {
            const int col = n0 + c * 16 + l16;
            v2f bfrag;
            bfrag.x = W[(kk)     * WH_COLS + col];
            bfrag.y = W[(kk + 1) * WH_COLS + col];
            acc[c] = __builtin_amdgcn_wmma_f32_16x16x4_f32(
                false, afrag, false, bfrag, (short)0, acc[c], false, false);
        }
    }

    // D layout: VGPR r -> M = r (lanes 0-15) / r+8 (lanes 16-31), N = lane%16
    #pragma unroll
    for (int c = 0; c < 4; ++c) {
        #pragma unroll
        for (int r = 0; r < 8; ++r) {
            const int rr = m0 + r + 8 * hlf;
            const int cc = n0 + c * 16 + l16;
            Wh[rr * WH_COLS + cc] = acc[c][r];
        }
    }
}

// ---------------------------------------------------------------------------
// Kernel 2: per (n,h): f_i = <Wh[n,h,:], a_dst[h]>, f_j = <Wh[n,h,:], a_src[h]>
// plus the 4 exponentials used by the branch-factorized softmax weight.
// Aux arrays are head-major: arr[h*N + n].
// ---------------------------------------------------------------------------
__global__ __launch_bounds__(256) void feat_kernel(const float* __restrict__ Wh,
                                                   const float* __restrict__ avec,
                                                   float* __restrict__ fi,
                                                   float* __restrict__ fj,
                                                   float* __restrict__ E1,
                                                   float* __restrict__ E2,
                                                   float* __restrict__ A1,
                                                   float* __restrict__ A2) {
    const int t = blockIdx.x * blockDim.x + threadIdx.x;
    if (t >= N_NODES * HEADS) return;
    const int n = t >> 2;
    const int h = t & 3;
    const float* w  = Wh + n * WH_COLS + h * OUT_DIM;
    const float* ad = avec + h * (2 * OUT_DIM);
    const float* as = ad + OUT_DIM;
    float si = 0.f, sj = 0.f;
    #pragma unroll 8
    for (int d = 0; d < OUT_DIM; ++d) {
        const float v = w[d];
        si += v * ad[d];
        sj += v * as[d];
    }
    const int idx = h * N_NODES + n;
    fi[idx] = si;
    fj[idx] = sj;
    E1[idx] = __expf(sj);          // exp(f_j)
    E2[idx] = __expf(0.2f * sj);   // exp(0.2 f_j)  (leaky slope branch)
    A1[idx] = __expf(si);          // exp(f_i)
    A2[idx] = __expf(0.2f * si);
}

// ---------------------------------------------------------------------------
// Kernel 3: invZ[i,h] = 1 / sum_j w_ij, w_ij = (f_i+f_j>=0) ? A1*E1[j] : A2*E2[j]
// One wave per (i,h); lanes stride j. No transcendentals in the N^2 loop.
// ---------------------------------------------------------------------------
__global__ __launch_bounds__(128) void z_kernel(const float* __restrict__ fi,
                                                const float* __restrict__ fj,
                                                const float* __restrict__ E1,
                                                const float* __restrict__ E2,
                                                const float* __restrict__ A1,
                                                const float* __restrict__ A2,
                                                float* __restrict__ invZ) {
    const int wave = (blockIdx.x * blockDim.x + threadIdx.x) >> 5;
    const int lane = threadIdx.x & 31;
    const int h = wave & 3;
    const int i = wave >> 2;
    const int base = h * N_NODES;
    const float fiv = fi[base + i];
    const float a1  = A1[base + i];
    const float a2  = A2[base + i];
    const float* fjh = fj + base;
    const float* e1h = E1 + base;
    const float* e2h = E2 + base;
    float s = 0.f;
    for (int j = lane; j < N_NODES; j += 32) {
        const float tt = fiv + fjh[j];
        s += (tt >= 0.f) ? a1 * e1h[j] : a2 * e2h[j];
    }
    s = waveReduceSum(s);
    if (lane == 0) invZ[base + i] = 1.0f / s;
}

// ---------------------------------------------------------------------------
// Kernel 4 (hot): one wave per (16-row i-tile, 256-col j-chunk).
//  - stages per-head f_j/E1/E2 chunk into LDS with GLOBAL_LOAD_ASYNC_TO_LDS_B128
//    (ASYNCcnt-tracked, no VGPR round-trip), s_wait_asynccnt before barrier
//  - per 4-j step: computes attn for all 4 heads (8 vals/lane), writes two
//    contiguous 16B NON-TEMPORAL stores ([i][j][h] layout => heads contiguous;
//    attn is a 268MB write-once stream -> keep it out of the 192MB L2 so Wh
//    and the aux arrays stay resident), and feeds the same values as WMMA
//    A-fragments: out += attn @ Wh.
//  - 16 v_wmma_f32_16x16x4_f32 per step (4 heads x 4 d-tiles).
//  - partial 16x(4x64) tiles combined via global_atomic_add_f32.
// ---------------------------------------------------------------------------
__global__ __launch_bounds__(32) void attn_out_kernel(const float* __restrict__ Wh,
                                                      const float* __restrict__ fi,
                                                      const float* __restrict__ fj,
                                                      const float* __restrict__ E1,
                                                      const float* __restrict__ E2,
                                                      const float* __restrict__ A1,
                                                      const float* __restrict__ A2,
                                                      const float* __restrict__ invZ,
                                                      float* __restrict__ out,
                                                      float* __restrict__ attn) {
    __shared__ float s_fj[HEADS][256];
    __shared__ float s_e1[HEADS][256];
    __shared__ float s_e2[HEADS][256];

    const int lane = threadIdx.x;
    const int hlf  = lane >> 4;
    const int l16  = lane & 15;
    const int it = blockIdx.x >> 4;       // 0..255  : i tile
    const int jc = blockIdx.x & 15;       // 0..15   : j chunk
    const int i0 = it * 16;
    const int jbase = jc * 256;

    // ---- async stage of the j-chunk of per-head arrays into LDS ----
    // Each lane copies 16B chunks: GVS mode (SADDR base + 32-bit byte offset),
    // VDST holds the LDS byte address. 24 async b128 ops total per lane-set.
    {
        const int q0 = lane * 4;          // float index within 256-float chunk
        #pragma unroll
        for (int h = 0; h < HEADS; ++h) {
            const unsigned gbase = (unsigned)((h * N_NODES + jbase) * sizeof(float));
            #pragma unroll
            for (int p = 0; p < 2; ++p) {
                const int q = q0 + p * 128;
                const unsigned goff = gbase + (unsigned)(q * sizeof(float));
                const unsigned l_fj = ldsOff(&s_fj[h][q]);
                const unsigned l_e1 = ldsOff(&s_e1[h][q]);
                const unsigned l_e2 = ldsOff(&s_e2[h][q]);
                asm volatile("global_load_async_to_lds_b128 %0, %1, %2"
                             :: "v"(l_fj), "v"(goff), "s"(fj) : "memory");
                asm volatile("global_load_async_to_lds_b128 %0, %1, %2"
                             :: "v"(l_e1), "v"(goff), "s"(E1) : "memory");
                asm volatile("global_load_async_to_lds_b128 %0, %1, %2"
                             :: "v"(l_e2), "v"(goff), "s"(E2) : "memory");
            }
        }
        asm volatile("s_wait_asynccnt 0" ::: "memory");
    }
    __syncthreads();

    const int i = i0 + l16;               // this lane's output row (both halves)
    float fiv[HEADS], a1v[HEADS], a2v[HEADS], izv[HEADS];
    #pragma unroll
    for (int h = 0; h < HEADS; ++h) {
        const int idx = h * N_NODES + i;
        fiv[h] = fi[idx];
        a1v[h] = A1[idx];
        a2v[h] = A2[idx];
        izv[h] = invZ[idx];
    }

    v8f acc[HEADS][4] = {};               // [head][d-tile], 16x16 each

    for (int js = 0; js < 256; js += 4) {
        const int jloc = js + 2 * hlf;    // this lane's K-pair (matches A-frag map)
        const int j = jbase + jloc;

        v2f afr[HEADS];
        v4f at0, at1;
        #pragma unroll
        for (int h = 0; h < HEADS; ++h) {
            const float t0 = fiv[h] + s_fj[h][jloc];
            const float t1 = fiv[h] + s_fj[h][jloc + 1];
            const float w0 = (t0 >= 0.f) ? a1v[h] * s_e1[h][jloc]     : a2v[h] * s_e2[h][jloc];
            const float w1 = (t1 >= 0.f) ? a1v[h] * s_e1[h][jloc + 1] : a2v[h] * s_e2[h][jloc + 1];
            const float v0 = w0 * izv[h];
            const float v1 = w1 * izv[h];
            afr[h].x = v0;
            afr[h].y = v1;
            at0[h] = v0;
            at1[h] = v1;
        }

        // attn[i][j][h]: heads contiguous -> two 16B streaming (NT) stores/lane
        float* arow = attn + (size_t)i * (N_NODES * HEADS);
        __builtin_nontemporal_store(at0, (v4f*)(arow + (size_t)j * HEADS));
        __builtin_nontemporal_store(at1, (v4f*)(arow + (size_t)(j + 1) * HEADS));

        // out_tile += attn_frag @ Wh rows j..j+3 (per head, 4 d-tiles of 16)
        #pragma unroll
        for (int h = 0; h < HEADS; ++h) {
            #pragma unroll
            for (int c = 0; c < 4; ++c) {
                const int col = h * OUT_DIM + c * 16 + l16;
                v2f bfrag;
                bfrag.x = Wh[(size_t)j * WH_COLS + col];
                bfrag.y = Wh[(size_t)(j + 1) * WH_COLS + col];
                acc[h][c] = __builtin_amdgcn_wmma_f32_16x16x4_f32(
                    false, afr[h], false, bfrag, (short)0, acc[h][c], false, false);
            }
        }
    }

    // combine partial tiles across j-chunks
    #pragma unroll
    for (int h = 0; h < HEADS; ++h)
        #pragma unroll
        for (int c = 0; c < 4; ++c)
            #pragma unroll
            for (int r = 0; r < 8; ++r) {
                const int row = i0 + r + 8 * hlf;
                const int col = h * OUT_DIM + c * 16 + l16;
                atomicAdd(&out[row * WH_COLS + col], acc[h][c][r]);
            }
}

// ---------------------------------------------------------------------------
extern "C" void kernel_launch(void* const* d_in, const int* in_sizes, int n_in,
                              void* d_out, int out_size, void* d_ws, size_t ws_size,
                              hipStream_t stream) {
    const float* hmat = (const float*)d_in[0];   // [4096, 256]
    const float* W    = (const float*)d_in[1];   // [256, 256]
    const float* avec = (const float*)d_in[2];   // [4, 128]

    float* out  = (float*)d_out;                 // [4096, 256]
    float* attn = out + (size_t)N_NODES * WH_COLS;  // [4096, 4096, 4]

    // workspace layout
    float* Wh   = (float*)d_ws;                  // 4096*256
    float* fi   = Wh + (size_t)N_NODES * WH_COLS;
    float* fj   = fi + (size_t)HEADS * N_NODES;
    float* E1   = fj + (size_t)HEADS * N_NODES;
    float* E2   = E1 + (size_t)HEADS * N_NODES;
    float* A1   = E2 + (size_t)HEADS * N_NODES;
    float* A2   = A1 + (size_t)HEADS * N_NODES;
    float* invZ = A2 + (size_t)HEADS * N_NODES;

    hipMemsetAsync(out, 0, (size_t)N_NODES * WH_COLS * sizeof(float), stream);

    // 1) Wh = h @ W : 256 row-tiles x 4 col-groups, one wave each
    gemm_wh<<<1024, 32, 0, stream>>>(hmat, W, Wh);

    // 2) features + exponentials
    feat_kernel<<<(N_NODES * HEADS + 255) / 256, 256, 0, stream>>>(
        Wh, avec, fi, fj, E1, E2, A1, A2);

    // 3) softmax denominators (one wave per (i,h))
    z_kernel<<<(N_NODES * HEADS) / 4, 128, 0, stream>>>(
        fi, fj, E1, E2, A1, A2, invZ);

    // 4) fused attn write + out accumulation (256 i-tiles x 16 j-chunks)
    attn_out_kernel<<<256 * 16, 32, 0, stream>>>(
        Wh, fi, fj, E1, E2, A1, A2, invZ, out, attn);
}